// DeconvLayer_61787399520298
// MI455X (gfx1250) — compile-verified
//
#include <hip/hip_runtime.h>

typedef __attribute__((ext_vector_type(2))) float v2f;
typedef __attribute__((ext_vector_type(4))) float v4f;
typedef __attribute__((ext_vector_type(8))) float v8f;

#define Bq 256
#define Nq 131072
#define Lq 1024
#define Kq (Nq / Lq)   // 128 blocks per row

// workspace layout (float offsets)
#define WS_H  0                       // L*8 homogeneous responses
#define WS_M8 (Lq * 8)                // 8x8 block transition
#define WS_E  (WS_M8 + 64)            // B*K*8 block end states (zero-init)
#define WS_S  (WS_E + Bq * Kq * 8)    // B*K*8 incoming states

// ---------------- k1: homogeneous responses H and transition M8 ----------------
__global__ void k1_homog(const float* __restrict__ w, float* __restrict__ ws) {
    int j = threadIdx.x;
    if (j >= 8) return;
    float wr[8];
#pragma unroll
    for (int t = 0; t < 8; ++t) wr[t] = w[t];
    float c[8];
#pragma unroll
    for (int t = 0; t < 8; ++t) c[t] = (t == j) ? 1.0f : 0.0f;
    float* H = ws + WS_H;
    for (int i = 0; i < Lq; ++i) {
        float v = 0.0f;
#pragma unroll
        for (int t = 0; t < 8; ++t) v = fmaf(wr[t], c[t], v);
        H[i * 8 + j] = v;
#pragma unroll
        for (int t = 7; t > 0; --t) c[t] = c[t - 1];
        c[0] = v;
    }
    float* M8 = ws + WS_M8;
#pragma unroll
    for (int r = 0; r < 8; ++r) M8[r * 8 + j] = c[r];   // s_out = E + M8 @ s_in
}

// ---------------- k2: per-block particular solution (zero initial carry) ------
__global__ void k2_particular(const float* __restrict__ x, const float* __restrict__ w,
                              float* __restrict__ vp, float* __restrict__ ws) {
    const int b = threadIdx.x;   // 0..255
    const int k = blockIdx.x;    // 0..127
    float wr[8];
#pragma unroll
    for (int t = 0; t < 8; ++t) wr[t] = w[t];
    float c[8] = {0, 0, 0, 0, 0, 0, 0, 0};
    const float* xp = x  + (size_t)b * Nq + (size_t)k * Lq;
    float*       op = vp + (size_t)b * Nq + (size_t)k * Lq;
    const bool head = (k == 0);
    for (int i = 0; i < Lq; i += 8) {
        v4f x0 = __builtin_nontemporal_load((const v4f*)(xp + i));
        v4f x1 = __builtin_nontemporal_load((const v4f*)(xp + i + 4));
        float xv[8], ov[8];
#pragma unroll
        for (int u = 0; u < 4; ++u) { xv[u] = x0[u]; xv[u + 4] = x1[u]; }
#pragma unroll
        for (int u = 0; u < 8; ++u) {
            float v = xv[u];
#pragma unroll
            for (int t = 0; t < 8; ++t) v = fmaf(wr[t], c[t], v);
            if (head && (i + u) < 8) v = 0.0f;   // v[i]=0 for i<8 (reference rule)
            ov[u] = v;
#pragma unroll
            for (int t = 7; t > 0; --t) c[t] = c[t - 1];
            c[0] = v;
        }
        v4f o0, o1;
#pragma unroll
        for (int u = 0; u < 4; ++u) { o0[u] = ov[u]; o1[u] = ov[u + 4]; }
        *(v4f*)(op + i)     = o0;
        *(v4f*)(op + i + 4) = o1;
    }
    float* E = ws + WS_E + ((size_t)b * Kq + k) * 8;
#pragma unroll
    for (int t = 0; t < 8; ++t) E[t] = c[t];
}

// ---------------- k3: serial scan over blocks: s' = E + M8 @ s ----------------
__global__ void k3_scan(float* __restrict__ ws) {
    const int t = blockIdx.x * blockDim.x + threadIdx.x;  // 0..2047
    const int b = t >> 3;
    const int j = t & 7;
    const float* M8 = ws + WS_M8;
    float m[8];
#pragma unroll
    for (int u = 0; u < 8; ++u) m[u] = M8[j * 8 + u];
    const float* E = ws + WS_E;
    float*       S = ws + WS_S;
    float s = 0.0f;                                       // block 0 incoming state
    for (int k = 0; k < Kq; ++k) {
        const size_t base = ((size_t)b * Kq + k) * 8;
        S[base + j] = s;
        float acc = E[base + j];
#pragma unroll
        for (int u = 0; u < 8; ++u) {
            float sv = __shfl(s, u, 8);                   // gather s[u] within 8-lane group
            acc = fmaf(m[u], sv, acc);
        }
        s = acc;
    }
}

// ---------------- k4: WMMA fixup: out = vp + H(16x8) @ S(8x16) ----------------
__global__ void k4_fix(float* __restrict__ out, const float* __restrict__ ws) {
    const int wid  = blockIdx.x * (blockDim.x >> 5) + (threadIdx.x >> 5);
    const int lane = threadIdx.x & 31;
    const int l15  = lane & 15;
    const int hi   = lane >> 4;
    const int tb   = wid & 15;       // row group   (16 rows)
    const int tp   = wid >> 4;       // position chunk (16 positions)
    const int b0   = tb * 16;
    const int p0   = tp * 16;
    const int k    = p0 / Lq;
    const int pl   = p0 % Lq;

    const float* H = ws + WS_H;
    const float* S = ws + WS_S;

    // A operand (16x4 f32): lane<16 holds {K=k0,k0+1}, lane>=16 holds {K=k0+2,k0+3}
    const float* hrow = H + (size_t)(pl + l15) * 8 + 2 * hi;
    v2f a0 = *(const v2f*)(hrow);        // k0 = 0
    v2f a1 = *(const v2f*)(hrow + 4);    // k0 = 4
    // B operand (4x16 f32): N across lanes 0-15, K split like A
    const float* srow = S + ((size_t)(b0 + l15) * Kq + k) * 8 + 2 * hi;
    v2f bb0 = *(const v2f*)(srow);
    v2f bb1 = *(const v2f*)(srow + 4);

    // C operand = vp tile (currently in d_out): VGPR r -> position p0 + r + 8*hi
    float* cp = out + (size_t)(b0 + l15) * Nq + p0 + hi * 8;
    v4f c0 = *(const v4f*)(cp);
    v4f c1 = *(const v4f*)(cp + 4);
    v8f cc;
#pragma unroll
    for (int u = 0; u < 4; ++u) { cc[u] = c0[u]; cc[u + 4] = c1[u]; }

    v8f d = __builtin_amdgcn_wmma_f32_16x16x4_f32(false, a0, false, bb0, (short)0, cc, false, false);
    d     = __builtin_amdgcn_wmma_f32_16x16x4_f32(false, a1, false, bb1, (short)0, d,  false, false);

    v4f d0, d1;
#pragma unroll
    for (int u = 0; u < 4; ++u) { d0[u] = d[u]; d1[u] = d[u + 4]; }
    *(v4f*)(cp)     = d0;
    *(v4f*)(cp + 4) = d1;
}

extern "C" void kernel_launch(void* const* d_in, const int* in_sizes, int n_in,
                              void* d_out, int out_size, void* d_ws, size_t ws_size,
                              hipStream_t stream) {
    const float* x = (const float*)d_in[0];   // [B, N] f32
    const float* w = (const float*)d_in[1];   // [8] f32
    float* out = (float*)d_out;               // [B, N] f32 (also holds vp between k2 and k4)
    float* ws  = (float*)d_ws;                // needs ~2.13 MB

    k1_homog<<<1, 32, 0, stream>>>(w, ws);
    k2_particular<<<Kq, Bq, 0, stream>>>(x, w, out, ws);
    k3_scan<<<(Bq * 8) / 256, 256, 0, stream>>>(ws);
    const int tiles = (Bq / 16) * (Nq / 16);            // 131072 waves
    k4_fix<<<tiles / 8, 256, 0, stream>>>(out, ws);
}